// RGAT_6399501271542
// MI455X (gfx1250) — compile-verified
//
#include <hip/hip_runtime.h>
#include <math.h>

typedef __attribute__((ext_vector_type(2))) float v2f;
typedef __attribute__((ext_vector_type(8))) float v8f;

// ---------------------------------------------------------------------------
// Kernel 0: zero workspace accumulators (deg, agg1, agg2, pool_sum, pool_cnt)
// ---------------------------------------------------------------------------
__global__ void zero_ws_kernel(float* __restrict__ p, long n) {
    long i = (long)blockIdx.x * blockDim.x + threadIdx.x;
    if (i < n) p[i] = 0.0f;
}

// ---------------------------------------------------------------------------
// Kernel 1: conv1 edge scatter: agg1[dst] += x[src] (F_IN=2), deg[dst] += 1
// One thread per edge. f32 atomics resolve in L2.
// ---------------------------------------------------------------------------
__global__ void edge_scatter1_kernel(const float* __restrict__ x,
                                     const int* __restrict__ esrc,
                                     const int* __restrict__ edst,
                                     float* __restrict__ agg1,
                                     float* __restrict__ deg,
                                     int E) {
    int e = blockIdx.x * blockDim.x + threadIdx.x;
    if (e >= E) return;
    // speculative stream prefetch of the edge arrays (global_prefetch_b8)
    __builtin_prefetch(&esrc[e + 4096], 0, 1);
    __builtin_prefetch(&edst[e + 4096], 0, 1);
    int s = esrc[e];
    int d = edst[e];
    atomicAdd(&agg1[2 * d + 0], x[2 * s + 0]);
    atomicAdd(&agg1[2 * d + 1], x[2 * s + 1]);
    atomicAdd(&deg[d], 1.0f);
}

// ---------------------------------------------------------------------------
// Kernel 2: conv1 node epilogue + relu:
//   h1[i][j] = relu( mean(i)@W1_l[:,j] + b1[j] + x[i]@W1_r[:,j] )
// Thread per (node, j). Also counts nodes per graph (j==0 lane).
// ---------------------------------------------------------------------------
__global__ void node_update1_kernel(const float* __restrict__ x,
                                    const float* __restrict__ agg1,
                                    const float* __restrict__ deg,
                                    const float* __restrict__ W1l,  // [2,16]
                                    const float* __restrict__ b1,   // [16]
                                    const float* __restrict__ W1r,  // [2,16]
                                    const int* __restrict__ batch,
                                    float* __restrict__ h1,         // [N,16]
                                    float* __restrict__ pool_cnt,   // [G]
                                    int N) {
    int idx = blockIdx.x * blockDim.x + threadIdx.x;
    if (idx >= N * 16) return;
    int i = idx >> 4;
    int j = idx & 15;
    float dg = fmaxf(deg[i], 1.0f);
    float m0 = agg1[2 * i + 0] / dg;
    float m1 = agg1[2 * i + 1] / dg;
    float h = m0 * W1l[j] + m1 * W1l[16 + j] + b1[j]
            + x[2 * i + 0] * W1r[j] + x[2 * i + 1] * W1r[16 + j];
    h1[idx] = fmaxf(h, 0.0f);
    if (j == 0) atomicAdd(&pool_cnt[batch[i]], 1.0f);
}

// ---------------------------------------------------------------------------
// Kernel 3: conv2 edge scatter: agg2[dst][0:16] += h1[src][0:16]
// 16 consecutive lanes per edge -> coalesced gather of h1[src].
// ---------------------------------------------------------------------------
__global__ void edge_scatter2_kernel(const float* __restrict__ h1,
                                     const int* __restrict__ esrc,
                                     const int* __restrict__ edst,
                                     float* __restrict__ agg2,
                                     long total) {
    long idx = (long)blockIdx.x * blockDim.x + threadIdx.x;
    if (idx >= total) return;
    int e = (int)(idx >> 4);
    int j = (int)(idx & 15);
    int s = esrc[e];
    int d = edst[e];
    atomicAdd(&agg2[16 * d + j], h1[16 * s + j]);
}

// ---------------------------------------------------------------------------
// Kernel 4: conv2 dense part via WMMA, one wave (32 lanes) per 16-node tile.
//   C = relu( Mean@W2_l + H1@W2_r + b2 )   (16x16 f32, exact fp32 semantics)
// then per-node out = C_row . W_end + b_end, atomic-added into pool_sum[g].
//
// V_WMMA_F32_16X16X4_F32 operand layouts (ISA 7.12.2, 32-bit):
//   A (16x4): lanes 0-15 = M, VGPR0 holds K={k0 | k0+2}, VGPR1 K={k0+1 | k0+3}
//             (halves of the wave take K offset +2)
//   B (4x16): N striped across lanes within a VGPR, mirrored K interleave
//   C/D (16x16): VGPR r = rows {r, r+8} across lane halves, N = lane&15
// ---------------------------------------------------------------------------
__global__ void __launch_bounds__(32)
sage2_wmma_tile_kernel(const float* __restrict__ agg2,
                       const float* __restrict__ deg,
                       const float* __restrict__ h1,
                       const float* __restrict__ W2l,   // [16,16]
                       const float* __restrict__ b2,    // [16]
                       const float* __restrict__ W2r,   // [16,16]
                       const float* __restrict__ Wend,  // [16,1]
                       const float* __restrict__ bend,  // [1]
                       const int* __restrict__ batch,
                       float* __restrict__ pool_sum,    // [G]
                       int N) {
    __shared__ float sA[16][16];    // mean tile
    __shared__ float sH[16][16];    // h1 tile
    __shared__ float sWl[16][16];
    __shared__ float sWr[16][16];
    __shared__ float sOut[16][16];
    __shared__ float sWend[16];

    const int lane = threadIdx.x;   // 0..31, wave32
    const int nb = blockIdx.x * 16; // tile base node

    // Stage tiles + weights into LDS (8 elements per lane per array).
#pragma unroll
    for (int r = 0; r < 8; ++r) {
        int idx = r * 32 + lane;        // 0..255
        int m = idx >> 4;
        int k = idx & 15;
        int node = nb + m;
        float mv = 0.0f, hv = 0.0f;
        if (node < N) {
            float dg = fmaxf(deg[node], 1.0f);
            mv = agg2[16 * node + k] / dg;
            hv = h1[16 * node + k];
        }
        sA[m][k] = mv;
        sH[m][k] = hv;
        sWl[m][k] = W2l[idx];
        sWr[m][k] = W2r[idx];
    }
    if (lane < 16) sWend[lane] = Wend[lane];
    __syncthreads();

    const int ln = lane & 15;       // = M for A loads, = N for B/C
    const int hi = lane >> 4;       // wave half selects K offset +2 / row +8

    v8f c;
    float bn = b2[ln];
#pragma unroll
    for (int r = 0; r < 8; ++r) c[r] = bn;   // seed accumulator with bias

    // C += Mean @ W2_l   (K=16 as 4 steps of K=4)
#pragma unroll
    for (int k0 = 0; k0 < 16; k0 += 4) {
        int kb = k0 + (hi << 1);
        v2f a, b;
        a.x = sA[ln][kb];       a.y = sA[ln][kb + 1];
        b.x = sWl[kb][ln];      b.y = sWl[kb + 1][ln];
        c = __builtin_amdgcn_wmma_f32_16x16x4_f32(
                false, a, false, b, (short)0, c, false, false);
    }
    // C += H1 @ W2_r
#pragma unroll
    for (int k0 = 0; k0 < 16; k0 += 4) {
        int kb = k0 + (hi << 1);
        v2f a, b;
        a.x = sH[ln][kb];       a.y = sH[ln][kb + 1];
        b.x = sWr[kb][ln];      b.y = sWr[kb + 1][ln];
        c = __builtin_amdgcn_wmma_f32_16x16x4_f32(
                false, a, false, b, (short)0, c, false, false);
    }

    // relu + spill D to LDS per C/D layout (row = r + 8*hi, col = ln)
#pragma unroll
    for (int r = 0; r < 8; ++r)
        sOut[r + 8 * hi][ln] = fmaxf(c[r], 0.0f);
    __syncthreads();

    // Final linear: out[node] = row . W_end + b_end; pool by graph id.
    if (lane < 16) {
        int node = nb + lane;
        if (node < N) {
            float s = 0.0f;
#pragma unroll
            for (int j = 0; j < 16; ++j) s += sOut[lane][j] * sWend[j];
            s += bend[0];
            atomicAdd(&pool_sum[batch[node]], s);
        }
    }
}

// ---------------------------------------------------------------------------
// Kernel 5: pooled mean + sigmoid -> d_out [G]
// ---------------------------------------------------------------------------
__global__ void pool_sigmoid_kernel(const float* __restrict__ pool_sum,
                                    const float* __restrict__ pool_cnt,
                                    float* __restrict__ out, int G) {
    int g = blockIdx.x * blockDim.x + threadIdx.x;
    if (g >= G) return;
    float m = pool_sum[g] / fmaxf(pool_cnt[g], 1.0f);
    out[g] = 1.0f / (1.0f + expf(-m));
}

// ---------------------------------------------------------------------------
extern "C" void kernel_launch(void* const* d_in, const int* in_sizes, int n_in,
                              void* d_out, int out_size, void* d_ws, size_t ws_size,
                              hipStream_t stream) {
    const float* x    = (const float*)d_in[0];   // [N,2]
    const int*   eidx = (const int*)d_in[1];     // [2,E]
    const int*   batch= (const int*)d_in[2];     // [N]
    const float* W1l  = (const float*)d_in[3];   // [2,16]
    const float* b1   = (const float*)d_in[4];   // [16]
    const float* W1r  = (const float*)d_in[5];   // [2,16]
    const float* W2l  = (const float*)d_in[6];   // [16,16]
    const float* b2   = (const float*)d_in[7];   // [16]
    const float* W2r  = (const float*)d_in[8];   // [16,16]
    const float* Wend = (const float*)d_in[9];   // [16,1]
    const float* bend = (const float*)d_in[10];  // [1]

    const int N = in_sizes[0] / 2;
    const int E = in_sizes[1] / 2;
    const int G = out_size;
    const int* esrc = eidx;
    const int* edst = eidx + E;

    // Workspace layout (floats). Zeroed region is contiguous up front.
    float* ws   = (float*)d_ws;
    float* deg  = ws;                     // [N]
    float* agg1 = deg  + (long)N;         // [2N]
    float* agg2 = agg1 + 2L * N;          // [16N]
    float* psum = agg2 + 16L * N;         // [G]
    float* pcnt = psum + G;               // [G]
    float* h1   = pcnt + G;               // [16N]  (fully overwritten, no zero)

    const long nzero = 19L * N + 2L * G;

    zero_ws_kernel<<<(unsigned)((nzero + 255) / 256), 256, 0, stream>>>(ws, nzero);

    edge_scatter1_kernel<<<(E + 255) / 256, 256, 0, stream>>>(
        x, esrc, edst, agg1, deg, E);

    node_update1_kernel<<<(N * 16 + 255) / 256, 256, 0, stream>>>(
        x, agg1, deg, W1l, b1, W1r, batch, h1, pcnt, N);

    const long tot2 = 16L * E;
    edge_scatter2_kernel<<<(unsigned)((tot2 + 255) / 256), 256, 0, stream>>>(
        h1, esrc, edst, agg2, tot2);

    sage2_wmma_tile_kernel<<<(N + 15) / 16, 32, 0, stream>>>(
        agg2, deg, h1, W2l, b2, W2r, Wend, bend, batch, psum, N);

    pool_sigmoid_kernel<<<(G + 63) / 64, 64, 0, stream>>>(psum, pcnt, (float*)d_out, G);
}